// GNN_LSTM_Classifier_63797444214835
// MI455X (gfx1250) — compile-verified
//
#include <hip/hip_runtime.h>
#include <cstddef>
#include <cstdint>

// ---------------------------------------------------------------------------
// Problem constants (fixed by the reference)
// ---------------------------------------------------------------------------
#define N_NODES 20000
#define N_EDGES 640000
#define F_IN    128
#define G_HID   128
#define L_HID   64
#define N_CLS   8
#define P_PATH  4
#define L_SEQ   16
#define D_PATH  129          // F_IN + 1
#define G4      (4 * L_HID)  // 256 gates

typedef float v2f __attribute__((ext_vector_type(2)));
typedef float v8f __attribute__((ext_vector_type(8)));

// CDNA5 native fp32 WMMA: D[16x16] = A[16x4] * B[4x16] + C  (wave32, exact fp32)
__device__ __forceinline__ v8f wmma4(v2f a, v2f b, v8f c) {
  return __builtin_amdgcn_wmma_f32_16x16x4_f32(
      /*neg_a=*/false, a, /*neg_b=*/false, b,
      /*c_mod=*/(short)0, c, /*reuse_a=*/false, /*reuse_b=*/false);
}

__device__ __forceinline__ float sigmoidf_(float x) {
  return 1.0f / (1.0f + expf(-x));
}

// ---------------------------------------------------------------------------
// utility: zero fp32 buffer
// ---------------------------------------------------------------------------
__global__ void zero_f32(float* __restrict__ p, int n) {
  int i = blockIdx.x * 256 + threadIdx.x;
  if (i < n) p[i] = 0.0f;
}

// ---------------------------------------------------------------------------
// GCN stage 1: XW = X[N,128] @ Wg[128,128]  (fp32 WMMA)
// one block = one 16-row tile; 8 waves each own one 16-col tile
// ---------------------------------------------------------------------------
__global__ void __launch_bounds__(256) gcn_xw(const float* __restrict__ X,
                                              const float* __restrict__ Wg,
                                              float* __restrict__ XW) {
  __shared__ float xa[16 * F_IN];  // 8 KB staged A tile
  const int row0 = blockIdx.x * 16;
  for (int i = threadIdx.x; i < 16 * F_IN; i += 256) {
    int r = i / F_IN, c = i % F_IN;
    xa[i] = X[(size_t)(row0 + r) * F_IN + c];
  }
  __syncthreads();

  const int wave = threadIdx.x >> 5;
  const int lane = threadIdx.x & 31;
  const int half = lane >> 4;
  const int n    = lane & 15;
  const int col0 = wave * 16;

  v8f acc = {0.f, 0.f, 0.f, 0.f, 0.f, 0.f, 0.f, 0.f};
  for (int k0 = 0; k0 < F_IN; k0 += 4) {
    const int k = k0 + 2 * half;
    v2f a, b;
    a.x = xa[n * F_IN + k];
    a.y = xa[n * F_IN + k + 1];
    b.x = Wg[(size_t)k * G_HID + col0 + n];
    b.y = Wg[(size_t)(k + 1) * G_HID + col0 + n];
    acc = wmma4(a, b, acc);
  }
#pragma unroll
  for (int r = 0; r < 8; ++r) {
    const int m = r + 8 * half;
    XW[(size_t)(row0 + m) * G_HID + col0 + n] = acc[r];
  }
}

// ---------------------------------------------------------------------------
// GCN stage 2: degree accumulation (edge weights into dst), then dinv
// ---------------------------------------------------------------------------
__global__ void deg_accum(const int* __restrict__ dst,
                          const float* __restrict__ ew,
                          float* __restrict__ deg, int ne) {
  int e = blockIdx.x * 256 + threadIdx.x;
  if (e < ne) atomicAdd(&deg[dst[e]], ew[e]);
}

__global__ void deg_to_dinv(float* __restrict__ deg, int n) {
  int i = blockIdx.x * 256 + threadIdx.x;
  if (i < n) {
    float d = deg[i] + 1.0f;  // self-loop weight 1
    deg[i] = (d > 0.0f) ? rsqrtf(d) : 0.0f;
  }
}

// ---------------------------------------------------------------------------
// GCN stage 3: normalized scatter-add, one wave per edge
// ---------------------------------------------------------------------------
__global__ void __launch_bounds__(256) gcn_agg(const int* __restrict__ src,
                                               const int* __restrict__ dst,
                                               const float* __restrict__ ew,
                                               const float* __restrict__ dinv,
                                               const float* __restrict__ XW,
                                               float* __restrict__ agg, int ne) {
  const int wid  = (blockIdx.x * 256 + threadIdx.x) >> 5;
  const int lane = threadIdx.x & 31;
  if (wid >= ne) return;
  const int s = src[wid], d = dst[wid];
  const float nrm = dinv[s] * ew[wid] * dinv[d];
  const float* xr = XW + (size_t)s * G_HID;
  float* orow = agg + (size_t)d * G_HID;
#pragma unroll
  for (int c = lane; c < G_HID; c += 32) atomicAdd(&orow[c], nrm * xr[c]);
}

// self-loop term + bias + ReLU, written in place into agg
__global__ void gcn_finalize(float* __restrict__ agg,
                             const float* __restrict__ XW,
                             const float* __restrict__ dinv,
                             const float* __restrict__ bg, int total) {
  int i = blockIdx.x * 256 + threadIdx.x;
  if (i < total) {
    int node = i / G_HID, c = i % G_HID;
    float di = dinv[node];
    float v = agg[i] + di * di * XW[i] + bg[c];
    agg[i] = (v > 0.0f) ? v : 0.0f;
  }
}

// ---------------------------------------------------------------------------
// BiLSTM: one block of 256 threads (8 waves) handles 16 sequences.
// Phase A (per direction): pre[256 rows x 256 gates] = x @ Wih^T + b  (WMMA, K=129)
// Phase B: 16-step recurrence, gates = pre[t] + h @ Whh^T (WMMA, K=64),
//          cell update in LDS. Final h -> global.
// ---------------------------------------------------------------------------
#define PLD 260   // pre row stride (floats)
#define GLD 260   // gates row stride
#define HLD 68    // h/c row stride
#define LSTM_SMEM_FLOATS (256 * PLD + 16 * GLD + 2 * 16 * HLD)

__global__ void __launch_bounds__(256, 1)
lstm_block(const float* __restrict__ Cmat,
           const float* __restrict__ Wih_f, const float* __restrict__ Whh_f,
           const float* __restrict__ bih_f, const float* __restrict__ bhh_f,
           const float* __restrict__ Wih_b, const float* __restrict__ Whh_b,
           const float* __restrict__ bih_b, const float* __restrict__ bhh_b,
           float* __restrict__ hfo, float* __restrict__ hbo) {
  extern __shared__ float smem[];
  float* pre   = smem;                  // 256 x PLD
  float* gates = pre + 256 * PLD;       // 16  x GLD
  float* hbuf  = gates + 16 * GLD;      // 16  x HLD
  float* cbuf  = hbuf + 16 * HLD;       // 16  x HLD

  const int blk  = blockIdx.x;          // sequences [blk*16, blk*16+16)
  const int tid  = threadIdx.x;
  const int wave = tid >> 5;
  const int lane = tid & 31;
  const int half = lane >> 4;
  const int n    = lane & 15;

  for (int dir = 0; dir < 2; ++dir) {
    const float* Wih = dir ? Wih_b : Wih_f;
    const float* Whh = dir ? Whh_b : Whh_f;
    const float* bih = dir ? bih_b : bih_f;
    const float* bhh = dir ? bhh_b : bhh_f;
    float* hout = dir ? hbo : hfo;

    // ---- Phase A: input projection into pre (M=256 rows, N=256 gates, K=129)
    for (int tile = wave; tile < 256; tile += 8) {
      const int row0 = (tile >> 4) << 4;
      const int col0 = (tile & 15) << 4;
      const int rr = row0 + n;  // this lane's A row (seq*16 + step)
      const float* xr = Cmat + ((size_t)blk * 256 + rr) * D_PATH;
      const int g = col0 + n;   // this lane's B column (gate index)
      const float* wr = Wih + (size_t)g * D_PATH;
      v8f acc = {0.f, 0.f, 0.f, 0.f, 0.f, 0.f, 0.f, 0.f};
      for (int k0 = 0; k0 < D_PATH; k0 += 4) {  // 33 steps, last padded
        const int k = k0 + 2 * half;
        v2f a, b;
        a.x = (k     < D_PATH) ? xr[k]     : 0.0f;
        a.y = (k + 1 < D_PATH) ? xr[k + 1] : 0.0f;
        b.x = (k     < D_PATH) ? wr[k]     : 0.0f;
        b.y = (k + 1 < D_PATH) ? wr[k + 1] : 0.0f;
        acc = wmma4(a, b, acc);
      }
      const float bs = bih[g] + bhh[g];
#pragma unroll
      for (int r = 0; r < 8; ++r)
        pre[(size_t)(row0 + r + 8 * half) * PLD + col0 + n] = acc[r] + bs;
    }
    // init h, c
    for (int i = tid; i < 16 * HLD; i += 256) { hbuf[i] = 0.0f; cbuf[i] = 0.0f; }
    __syncthreads();

    // ---- Phase B: recurrence (each wave owns 2 gate col-tiles)
    for (int ts = 0; ts < L_SEQ; ++ts) {
      const int t = dir ? (L_SEQ - 1 - ts) : ts;
      v8f acc[2];
#pragma unroll
      for (int q = 0; q < 2; ++q) {
        const int col0 = (wave * 2 + q) * 16;
#pragma unroll
        for (int r = 0; r < 8; ++r)  // seed accumulator with pre[seq, t, gate]
          acc[q][r] = pre[(size_t)((r + 8 * half) * L_SEQ + t) * PLD + col0 + n];
        const float* wr = Whh + (size_t)(col0 + n) * L_HID;
        for (int k0 = 0; k0 < L_HID; k0 += 4) {
          const int k = k0 + 2 * half;
          v2f a, b;
          a.x = hbuf[n * HLD + k];
          a.y = hbuf[n * HLD + k + 1];
          b.x = wr[k];
          b.y = wr[k + 1];
          acc[q] = wmma4(a, b, acc[q]);
        }
#pragma unroll
        for (int r = 0; r < 8; ++r)
          gates[(r + 8 * half) * GLD + col0 + n] = acc[q][r];
      }
      __syncthreads();
      // cell update: 16 seq x 64 hidden
      for (int i = tid; i < 16 * L_HID; i += 256) {
        const int s = i >> 6, hc = i & 63;
        const float ig = gates[s * GLD + hc];
        const float fg = gates[s * GLD + 64 + hc];
        const float gg = gates[s * GLD + 128 + hc];
        const float og = gates[s * GLD + 192 + hc];
        float cc = cbuf[s * HLD + hc];
        cc = sigmoidf_(fg) * cc + sigmoidf_(ig) * tanhf(gg);
        hbuf[s * HLD + hc] = sigmoidf_(og) * tanhf(cc);
        cbuf[s * HLD + hc] = cc;
      }
      __syncthreads();
    }

    // write final hidden state
    for (int i = tid; i < 16 * L_HID; i += 256) {
      const int s = i >> 6, hc = i & 63;
      hout[((size_t)blk * 16 + s) * L_HID + hc] = hbuf[s * HLD + hc];
    }
    __syncthreads();  // pre is reused by the next direction
  }
}

// ---------------------------------------------------------------------------
// mean over P=4 paths: hp[n, 0:64]=mean hf, hp[n, 64:128]=mean hb
// ---------------------------------------------------------------------------
__global__ void hp_mean(const float* __restrict__ hf,
                        const float* __restrict__ hb,
                        float* __restrict__ hp, int nn) {
  int i = blockIdx.x * 256 + threadIdx.x;
  if (i < nn * 2 * L_HID) {
    int node = i >> 7, c = i & 127;
    const float* s = (c < L_HID) ? hf : hb;
    int cc = c & 63;
    float acc = 0.0f;
#pragma unroll
    for (int p = 0; p < P_PATH; ++p)
      acc += s[((size_t)node * P_PATH + p) * L_HID + cc];
    hp[i] = 0.25f * acc;
  }
}

// ---------------------------------------------------------------------------
// fuse: out[n, j] = [h0 | hp] @ Wf + bf   (K=256, N=8)
// ---------------------------------------------------------------------------
__global__ void fuse(const float* __restrict__ h0, const float* __restrict__ hp,
                     const float* __restrict__ Wf, const float* __restrict__ bf,
                     float* __restrict__ out, int nn) {
  int i = blockIdx.x * 256 + threadIdx.x;
  if (i < nn * N_CLS) {
    const int node = i >> 3, j = i & 7;
    const float* a = h0 + (size_t)node * G_HID;
    const float* b = hp + (size_t)node * 2 * L_HID;
    float s = bf[j];
    for (int k = 0; k < G_HID; ++k)      s += a[k] * Wf[k * N_CLS + j];
    for (int k = 0; k < 2 * L_HID; ++k)  s += b[k] * Wf[(G_HID + k) * N_CLS + j];
    out[i] = s;
  }
}

// ---------------------------------------------------------------------------
// host launcher
// ---------------------------------------------------------------------------
extern "C" void kernel_launch(void* const* d_in, const int* in_sizes, int n_in,
                              void* d_out, int out_size, void* d_ws, size_t ws_size,
                              hipStream_t stream) {
  (void)in_sizes; (void)n_in; (void)out_size; (void)ws_size;
  const float* X     = (const float*)d_in[0];
  const int*   ei    = (const int*)d_in[1];      // [2, E] flat: src then dst
  const float* ea    = (const float*)d_in[2];
  const float* C     = (const float*)d_in[3];
  const float* Wg    = (const float*)d_in[4];
  const float* bg    = (const float*)d_in[5];
  const float* Wih_f = (const float*)d_in[6];
  const float* Whh_f = (const float*)d_in[7];
  const float* bih_f = (const float*)d_in[8];
  const float* bhh_f = (const float*)d_in[9];
  const float* Wih_b = (const float*)d_in[10];
  const float* Whh_b = (const float*)d_in[11];
  const float* bih_b = (const float*)d_in[12];
  const float* bhh_b = (const float*)d_in[13];
  const float* Wf    = (const float*)d_in[14];
  const float* bf    = (const float*)d_in[15];
  float* out = (float*)d_out;

  // workspace layout (floats)
  float* ws   = (float*)d_ws;
  float* xw   = ws;                                   // [N, 128], later reused as hp
  float* agg  = xw + (size_t)N_NODES * G_HID;         // [N, 128] accum -> h0 in place
  float* dinv = agg + (size_t)N_NODES * G_HID;        // [N] (deg accum -> dinv)
  float* hf   = dinv + 20480;                         // [N*P, 64]
  float* hb   = hf + (size_t)N_NODES * P_PATH * L_HID;

  const int* src = ei;
  const int* dst = ei + N_EDGES;

  // zero agg + deg (contiguous)
  const int zc = N_NODES * G_HID + 20480;
  zero_f32<<<(zc + 255) / 256, 256, 0, stream>>>(agg, zc);

  // GCN
  gcn_xw<<<N_NODES / 16, 256, 0, stream>>>(X, Wg, xw);
  deg_accum<<<(N_EDGES + 255) / 256, 256, 0, stream>>>(dst, ea, dinv, N_EDGES);
  deg_to_dinv<<<(N_NODES + 255) / 256, 256, 0, stream>>>(dinv, N_NODES);
  gcn_agg<<<N_EDGES / 8, 256, 0, stream>>>(src, dst, ea, dinv, xw, agg, N_EDGES);
  gcn_finalize<<<(N_NODES * G_HID + 255) / 256, 256, 0, stream>>>(agg, xw, dinv, bg,
                                                                  N_NODES * G_HID);

  // BiLSTM
  const size_t smem_bytes = (size_t)LSTM_SMEM_FLOATS * sizeof(float);
  (void)hipFuncSetAttribute(reinterpret_cast<const void*>(lstm_block),
                            hipFuncAttributeMaxDynamicSharedMemorySize,
                            (int)smem_bytes);
  lstm_block<<<N_NODES * P_PATH / 16, 256, smem_bytes, stream>>>(
      C, Wih_f, Whh_f, bih_f, bhh_f, Wih_b, Whh_b, bih_b, bhh_b, hf, hb);

  // mean over paths (hp reuses xw, which is dead after gcn_finalize)
  hp_mean<<<(N_NODES * 2 * L_HID + 255) / 256, 256, 0, stream>>>(hf, hb, xw, N_NODES);

  // fuse linear (h0 lives in agg)
  fuse<<<(N_NODES * N_CLS + 255) / 256, 256, 0, stream>>>(agg, xw, Wf, bf, out, N_NODES);
}